// Kalman_77790447665179
// MI455X (gfx1250) — compile-verified
//
#include <hip/hip_runtime.h>
#include <hip/hip_bf16.h>
#include <math.h>

// Problem constants (match reference)
#define XD 16
#define YD 8
#define UD 4
#define NB 2048
#define NT 256

// Flat output offsets (floats): mu_seq, v_seq, logp, y_seq, vy_seq
#define OFF_MU   ((size_t)0)
#define OFF_V    ((size_t)8388608)          // 2048*256*16
#define OFF_LOGP ((size_t)142606336)        // OFF_V + 2048*256*256
#define OFF_Y    ((size_t)142606337)
#define OFF_VY   ((size_t)146800641)        // OFF_Y + 2048*256*8

// Workspace offsets (floats)
#define WS_V     ((size_t)0)        // 256*256
#define WS_K     ((size_t)65536)    // 256*16*8
#define WS_S     ((size_t)98304)    // 256*8*8  (vy_t)
#define WS_LINV  ((size_t)114688)   // 256*8*8
#define WS_LD    ((size_t)131072)   // 256
#define WS_BSUM  ((size_t)131328)   // 2048

typedef __attribute__((ext_vector_type(2))) float v2f;
typedef __attribute__((ext_vector_type(8))) float v8f;

// D = A @ B for 16x16 f32 row-major matrices in LDS, via 4 chained
// v_wmma_f32_16x16x4_f32. One full wave32 required (EXEC all ones).
__device__ __forceinline__ void mm16(const float* __restrict__ Am,
                                     const float* __restrict__ Bm,
                                     float* __restrict__ Dm, int lane) {
  const int m = lane & 15;        // A-frag row / B-frag col / D col
  const int half = lane >> 4;     // K-half selector
  v8f c = {};
#pragma unroll
  for (int k0 = 0; k0 < 16; k0 += 4) {
    v2f a, b;
    a[0] = Am[m * 16 + k0 + 2 * half + 0];
    a[1] = Am[m * 16 + k0 + 2 * half + 1];
    b[0] = Bm[(k0 + 2 * half + 0) * 16 + m];
    b[1] = Bm[(k0 + 2 * half + 1) * 16 + m];
    c = __builtin_amdgcn_wmma_f32_16x16x4_f32(false, a, false, b,
                                              (short)0, c, false, false);
  }
#pragma unroll
  for (int r = 0; r < 8; ++r)
    Dm[(r + 8 * half) * 16 + m] = c[r];   // D row = r + 8*half, col = m
}

// ---------------- Kernel 1: shared covariance recursion (1 wave) ----------
__global__ __launch_bounds__(32) void Kalman_cov_kernel(
    const float* __restrict__ P0, const float* __restrict__ A,
    const float* __restrict__ C, const float* __restrict__ sigmax,
    const float* __restrict__ sigmay, float* __restrict__ ws) {
  __shared__ float sA[256], sAt[256], sCt[256], sCp[256], sQ[256], sR[64];
  __shared__ float sV[256], sP[256], sT[256], sPHt[256];
  __shared__ float sS[64], sL[64], sLinv[64], sKp[256], sKC[256];
  __shared__ float slogdet;
  const int lane = threadIdx.x;

  for (int i = lane; i < 256; i += 32) {
    int r = i >> 4, cc = i & 15;
    float a = A[i];
    sA[i] = a;
    sAt[cc * 16 + r] = a;                       // A^T
    sQ[i] = sigmax[i];
    sCt[i] = (cc < 8) ? C[cc * 16 + r] : 0.f;   // C^T padded to 16x16
    sCp[i] = (r < 8) ? C[r * 16 + cc] : 0.f;    // C padded (rows 8..15 = 0)
    sV[i] = P0[i];                              // batch-0 P0 (broadcast PSD)
  }
  for (int i = lane; i < 64; i += 32) sR[i] = sigmay[i];
  __syncthreads();

  float* Vws = ws + WS_V;
  float* Kws = ws + WS_K;
  float* Sws = ws + WS_S;
  float* Lws = ws + WS_LINV;
  float* LDws = ws + WS_LD;

  for (int t = 0; t < NT; ++t) {
    if (t == 0) {
      for (int i = lane; i < 256; i += 32) sP[i] = sV[i];       // P = P0
      __syncthreads();
    } else {
      mm16(sA, sV, sT, lane);  __syncthreads();                 // T = A V
      mm16(sT, sAt, sP, lane); __syncthreads();                 // P = T A^T
      for (int i = lane; i < 256; i += 32) sP[i] += sQ[i];      // + sigmax
      __syncthreads();
    }
    mm16(sP, sCt, sPHt, lane); __syncthreads();                 // PH^T (cols 0..7)
    mm16(sCp, sPHt, sT, lane); __syncthreads();                 // C P H^T
    for (int i = lane; i < 64; i += 32) {
      int r = i >> 3, cc = i & 7;
      sS[i] = sT[r * 16 + cc] + sR[i];                          // S = HPH^T + R
    }
    __syncthreads();

    // Cholesky of 8x8 S, logdet, and L^{-1} (serial on lane 0: tiny)
    if (lane == 0) {
      float Lm[8][8], Li[8][8];
#pragma unroll
      for (int i = 0; i < 8; ++i)
#pragma unroll
        for (int j = 0; j < 8; ++j) Lm[i][j] = sS[i * 8 + j];
      float ld = 0.f;
#pragma unroll
      for (int j = 0; j < 8; ++j) {
        float d = Lm[j][j];
#pragma unroll
        for (int k = 0; k < 8; ++k) if (k < j) d -= Lm[j][k] * Lm[j][k];
        d = sqrtf(d);
        Lm[j][j] = d;
        ld += logf(d);
        float inv = 1.f / d;
#pragma unroll
        for (int i = 0; i < 8; ++i) {
          if (i > j) {
            float v = Lm[i][j];
#pragma unroll
            for (int k = 0; k < 8; ++k) if (k < j) v -= Lm[i][k] * Lm[j][k];
            Lm[i][j] = v * inv;
          }
        }
      }
      slogdet = 2.f * ld;
#pragma unroll
      for (int i = 0; i < 8; ++i)
#pragma unroll
        for (int j = 0; j < 8; ++j) sL[i * 8 + j] = (j <= i) ? Lm[i][j] : 0.f;
#pragma unroll
      for (int i = 0; i < 8; ++i) {
        Li[i][i] = 1.f / Lm[i][i];
#pragma unroll
        for (int j = 0; j < 8; ++j) {
          if (j < i) {
            float s = 0.f;
#pragma unroll
            for (int k = 0; k < 8; ++k)
              if (k >= j && k < i) s += Lm[i][k] * Li[k][j];
            Li[i][j] = -s / Lm[i][i];
          }
        }
      }
#pragma unroll
      for (int i = 0; i < 8; ++i)
#pragma unroll
        for (int j = 0; j < 8; ++j) sLinv[i * 8 + j] = (j <= i) ? Li[i][j] : 0.f;
    }
    __syncthreads();

    // K = PH^T S^{-1}: lanes 0..15 each solve one row via fwd/back subst.
    if (lane < 16) {
      float rhs[8], yv[8], xk[8];
#pragma unroll
      for (int j = 0; j < 8; ++j) rhs[j] = sPHt[lane * 16 + j];
#pragma unroll
      for (int j = 0; j < 8; ++j) {
        float s = rhs[j];
#pragma unroll
        for (int k = 0; k < 8; ++k) if (k < j) s -= sL[j * 8 + k] * yv[k];
        yv[j] = s / sL[j * 8 + j];
      }
#pragma unroll
      for (int j = 7; j >= 0; --j) {
        float s = yv[j];
#pragma unroll
        for (int k = 0; k < 8; ++k) if (k > j) s -= sL[k * 8 + j] * xk[k];
        xk[j] = s / sL[j * 8 + j];
      }
#pragma unroll
      for (int j = 0; j < 8; ++j) sKp[lane * 16 + j] = xk[j];
    }
    for (int i = lane; i < 256; i += 32)
      if ((i & 15) >= 8) sKp[i] = 0.f;                          // zero-pad K
    __syncthreads();

    mm16(sKp, sCp, sKC, lane); __syncthreads();                 // K C
    for (int i = lane; i < 256; i += 32) {
      int r = i >> 4, cc = i & 15;
      sT[i] = ((r == cc) ? 1.f : 0.f) - sKC[i];                 // I - KC
    }
    __syncthreads();
    mm16(sT, sP, sV, lane); __syncthreads();                    // V = (I-KC)P

    for (int i = lane; i < 256; i += 32) Vws[t * 256 + i] = sV[i];
    for (int i = lane; i < 128; i += 32) {
      int r = i >> 3, cc = i & 7;
      Kws[t * 128 + i] = sKp[r * 16 + cc];
    }
    for (int i = lane; i < 64; i += 32) {
      Sws[t * 64 + i] = sS[i];
      Lws[t * 64 + i] = sLinv[i];
    }
    if (lane == 0) LDws[t] = slogdet;
    __syncthreads();
  }
}

// ------------- Kernel 2: broadcast v_seq / vy_seq (HBM-bound) -------------
__global__ __launch_bounds__(256) void Kalman_bcast_kernel(
    const float* __restrict__ ws, float* __restrict__ out) {
  const float4* V4 = (const float4*)(ws + WS_V);   // 16384 quads, L2-resident
  const float* Sws = ws + WS_S;                    // 16384 floats
  float4* v4 = (float4*)(out + OFF_V);
  float* vyo = out + OFF_VY;
  const size_t stride = (size_t)gridDim.x * blockDim.x;
  const size_t base = (size_t)blockIdx.x * blockDim.x + threadIdx.x;

  const size_t nv4 = (size_t)NB * 16384;           // v_seq quads
  for (size_t i = base; i < nv4; i += stride) v4[i] = V4[i & 16383];

  const size_t nvy = (size_t)NB * 16384;           // vy_seq floats (unaligned base)
  for (size_t i = base; i < nvy; i += stride) vyo[i] = Sws[i & 16383];
}

// ------------- Kernel 3: per-batch mean recursion + loglik partials -------
__global__ __launch_bounds__(256) void Kalman_mu_kernel(
    const float* __restrict__ y, const float* __restrict__ u,
    const float* __restrict__ mu0, const float* __restrict__ A,
    const float* __restrict__ B, const float* __restrict__ C,
    const float* __restrict__ ws, float* __restrict__ out,
    float* __restrict__ bsum) {
  __shared__ float sA[256], sB[64], sC[128];
  const int tid = threadIdx.x;
  if (tid < 256) sA[tid] = A[tid];
  if (tid < 64)  sB[tid] = B[tid];
  if (tid < 128) sC[tid] = C[tid];
  __syncthreads();

  const int b = blockIdx.x * blockDim.x + tid;     // 0..2047
  const float* Kws = ws + WS_K;
  const float* Lws = ws + WS_LINV;
  const float* LDws = ws + WS_LD;
  const float LOG2PI = 1.8378770664093453f;

  const float* yb = y + (size_t)b * NT * YD;
  const float* ub = u + (size_t)b * NT * UD;
  float* mu_out = out + OFF_MU + (size_t)b * NT * XD;
  float* y_out  = out + OFF_Y  + (size_t)b * NT * YD;

  float mu[XD];
  float acc = 0.f;

  for (int t = 0; t < NT; ++t) {
    float xp[XD];
    if (t == 0) {
#pragma unroll
      for (int i = 0; i < XD; ++i) xp[i] = mu0[(size_t)b * XD + i];
    } else {
#pragma unroll
      for (int i = 0; i < XD; ++i) {
        float s = 0.f;
#pragma unroll
        for (int j = 0; j < XD; ++j) s += sA[i * XD + j] * mu[j];
#pragma unroll
        for (int j = 0; j < UD; ++j) s += sB[i * UD + j] * ub[t * UD + j];
        xp[i] = s;
      }
    }
    const float* Kt = Kws + (size_t)t * 128;       // uniform → scalar loads
    const float* Lit = Lws + (size_t)t * 64;
    float yp[YD], dl[YD];
#pragma unroll
    for (int r = 0; r < YD; ++r) {
      float s = 0.f;
#pragma unroll
      for (int j = 0; j < XD; ++j) s += sC[r * XD + j] * xp[j];
      yp[r] = s;
      dl[r] = yb[t * YD + r] - s;
      y_out[t * YD + r] = s;
    }
#pragma unroll
    for (int i = 0; i < XD; ++i) {
      float s = xp[i];
#pragma unroll
      for (int r = 0; r < YD; ++r) s += Kt[i * YD + r] * dl[r];
      mu[i] = s;
      mu_out[t * XD + i] = s;
    }
    float zz = 0.f;
#pragma unroll
    for (int i = 0; i < YD; ++i) {
      float z = 0.f;
#pragma unroll
      for (int j = 0; j < YD; ++j) if (j <= i) z += Lit[i * YD + j] * dl[j];
      zz += z * z;
    }
    acc += (float)YD * LOG2PI + LDws[t] + zz;
  }
  bsum[b] = acc;
}

// ------------- Kernel 4: deterministic logp reduction ---------------------
__global__ __launch_bounds__(256) void Kalman_reduce_kernel(
    const float* __restrict__ bsum, float* __restrict__ out) {
  __shared__ float red[256];
  float s = 0.f;
  for (int i = threadIdx.x; i < NB; i += 256) s += bsum[i];
  red[threadIdx.x] = s;
  __syncthreads();
  for (int k = 128; k > 0; k >>= 1) {
    if (threadIdx.x < k) red[threadIdx.x] += red[threadIdx.x + k];
    __syncthreads();
  }
  if (threadIdx.x == 0)
    out[OFF_LOGP] = -0.5f * red[0] / (float)((size_t)NB * NT);
}

extern "C" void kernel_launch(void* const* d_in, const int* in_sizes, int n_in,
                              void* d_out, int out_size, void* d_ws,
                              size_t ws_size, hipStream_t stream) {
  (void)in_sizes; (void)n_in; (void)out_size; (void)ws_size;
  const float* y      = (const float*)d_in[0];
  const float* u      = (const float*)d_in[1];
  const float* mu0    = (const float*)d_in[2];
  const float* P0     = (const float*)d_in[3];
  const float* A      = (const float*)d_in[4];
  const float* B      = (const float*)d_in[5];
  const float* C      = (const float*)d_in[6];
  const float* sigmax = (const float*)d_in[7];
  const float* sigmay = (const float*)d_in[8];
  float* out = (float*)d_out;
  float* ws  = (float*)d_ws;

  Kalman_cov_kernel<<<1, 32, 0, stream>>>(P0, A, C, sigmax, sigmay, ws);
  Kalman_mu_kernel<<<NB / 256, 256, 0, stream>>>(y, u, mu0, A, B, C, ws, out,
                                                 ws + WS_BSUM);
  Kalman_bcast_kernel<<<8192, 256, 0, stream>>>(ws, out);
  Kalman_reduce_kernel<<<1, 256, 0, stream>>>(ws + WS_BSUM, out);
}